// RPN_62775241998751
// MI455X (gfx1250) — compile-verified
//
#include <hip/hip_runtime.h>
#include <math.h>

// Problem constants (from the reference)
#define MMAX 128
static __device__ constexpr float kIouLow     = 0.3f;
static __device__ constexpr float kIouHigh    = 0.6f;
static __device__ constexpr float kScaleClamp = 3.33220451017520392e+00f; // log(224/8)
static __device__ constexpr float kNeutral    = -100000000.0f;
static __device__ constexpr float kBig        = 3.0e38f;

// Padded LDS layout: 8 floats (32 B) per gt row:
//   [0..3] = x1,y1,x2,y2  -> one aligned ds_load_b128 in the hot loop
//   [4]    = cls          -> only read in the final gather
//   [5]    = area         -> one ds_load_b32 in the hot loop
//   [6..7] = pad
// Invalid rows (cls == -1) are poisoned: coords=(+BIG,+BIG,-BIG,-BIG), area=0
// => IoU value 0, which is output-equivalent to the reference's -1 masking
// (an invalid row can only win when q <= IOU_LOW, where the output is -1 anyway).
__global__ __launch_bounds__(256) void rpn_match_kernel(
    const float* __restrict__ anchors,   // N x 4
    const float* __restrict__ deltas,    // N x 4
    const float* __restrict__ gt,        // M x 5 (x1,y1,x2,y2,cls)
    float* __restrict__ out_prop,        // N x 4
    float* __restrict__ out_match,       // N x 5
    int N, int M)
{
    __shared__ __align__(16) float sBox[MMAX * 8];

    const int tid = threadIdx.x;
    const int nElem = M * 5;

    // --- CDNA5 async global->LDS fill (ASYNCcnt), scattering straight into the
    // padded layout: global byte e*4  ->  LDS byte (e/5)*32 + (e%5)*4.
    unsigned lds_base = (unsigned)(unsigned long long)(void*)sBox;
    for (int e = tid; e < nElem; e += (int)blockDim.x) {
        unsigned row = (unsigned)e / 5u;
        unsigned col = (unsigned)e - row * 5u;
        unsigned lds_addr = lds_base + row * 32u + col * 4u;
        unsigned voff     = (unsigned)e * 4u;
        // GVS mode: copies 4 bytes from the global address gt + voff to the LDS address in lds_addr
        asm volatile("global_load_async_to_lds_b32 %0, %1, %2"
                     :: "v"(lds_addr), "v"(voff), "s"(gt)
                     : "memory");
    }
#if __has_builtin(__builtin_amdgcn_s_wait_asynccnt)
    __builtin_amdgcn_s_wait_asynccnt(0);
#else
    asm volatile("s_wait_asynccnt 0x0" ::: "memory");
#endif
    __syncthreads();

    // Per-gt fixup pass: compute area; poison invalid rows so the hot loop
    // needs no validity logic.
    for (int j = tid; j < M; j += (int)blockDim.x) {
        float gx1 = sBox[j * 8 + 0], gy1 = sBox[j * 8 + 1];
        float gx2 = sBox[j * 8 + 2], gy2 = sBox[j * 8 + 3];
        bool valid = (sBox[j * 8 + 4] != -1.0f);
        sBox[j * 8 + 5] = valid ? (gx2 - gx1) * (gy2 - gy1) : 0.0f;
        if (!valid) {
            sBox[j * 8 + 0] =  kBig;
            sBox[j * 8 + 1] =  kBig;
            sBox[j * 8 + 2] = -kBig;
            sBox[j * 8 + 3] = -kBig;
        }
    }
    __syncthreads();

    const int i = (int)blockIdx.x * (int)blockDim.x + tid;
    if (i >= N) return;

    const float4 a = ((const float4*)anchors)[i];
    const float4 d = ((const float4*)deltas)[i];

    // ---- proposals = apply_deltas_to_anchors ----
    {
        float px = (a.x + a.z) * 0.5f;
        float py = (a.y + a.w) * 0.5f;
        float pw = (a.z - a.x) * 0.5f;
        float ph = (a.w - a.y) * 0.5f;
        float dw = fminf(d.z, kScaleClamp);
        float dh = fminf(d.w, kScaleClamp);
        float bx = px + pw * d.x;
        float by = py + ph * d.y;
        float bw = pw * expf(dw);
        float bh = ph * expf(dh);
        ((float4*)out_prop)[i] = make_float4(bx - bw, by - bh, bx + bw, by + bh);
    }

    // ---- row of pairwise IoU vs gt, tracked as exact fractions ----
    const float aarea = (a.z - a.x) * (a.w - a.y);

    float nb = -1.0f;  // best numerator
    float ub = 1.0f;   // best denominator (always > 0 once any row is seen)
    int bestj = 0;

    const float4* sRow = (const float4*)sBox;   // 2 float4 per padded row

    #pragma unroll 8
    for (int j = 0; j < M; ++j) {
        const float4 g = sRow[j * 2];           // ds_load_b128: x1,y1,x2,y2
        const float ga = sBox[j * 8 + 5];       // ds_load_b32:  area
        float x1 = fmaxf(a.x, g.x);
        float y1 = fmaxf(a.y, g.y);
        float x2 = fminf(a.z, g.z);
        float y2 = fminf(a.w, g.w);
        float inter = fmaxf(x2 - x1, 0.0f) * fmaxf(y2 - y1, 0.0f);
        float uni   = aarea + ga - inter;
        // q_j > q_best  <=>  inter*ub > nb*uni   (ub, uni > 0); strict '>'
        // keeps the FIRST maximum, matching jnp.argmax.
        bool better = (inter * ub > nb * uni);
        nb    = better ? inter : nb;
        ub    = better ? uni   : ub;
        bestj = better ? j     : bestj;
    }

    const float q = nb / ub;   // match_quality (single divide per anchor)

    const float4 gbest = sRow[bestj * 2];
    float m0 = gbest.x, m1 = gbest.y, m2 = gbest.z, m3 = gbest.w;
    float m4 = sBox[bestj * 8 + 4];
    if (q <= kIouLow) {
        m0 = m1 = m2 = m3 = m4 = -1.0f;
    } else if (q < kIouHigh) {
        m0 = m1 = m2 = m3 = m4 = kNeutral;
    }
    float* o = out_match + (size_t)i * 5;
    o[0] = m0; o[1] = m1; o[2] = m2; o[3] = m3; o[4] = m4;
}

extern "C" void kernel_launch(void* const* d_in, const int* in_sizes, int n_in,
                              void* d_out, int out_size, void* d_ws, size_t ws_size,
                              hipStream_t stream) {
    const float* anchors = (const float*)d_in[0];
    const float* deltas  = (const float*)d_in[1];
    const float* gt      = (const float*)d_in[2];

    const int N = in_sizes[0] / 4;
    const int M = in_sizes[2] / 5;   // == 128

    float* out_prop  = (float*)d_out;              // N*4 floats
    float* out_match = out_prop + (size_t)N * 4;   // N*5 floats

    const int block = 256;                         // 8 wave32 waves
    const int grid  = (N + block - 1) / block;
    rpn_match_kernel<<<grid, block, 0, stream>>>(anchors, deltas, gt,
                                                 out_prop, out_match, N, M);
}